// NodeLayer_73486890434702
// MI455X (gfx1250) — compile-verified
//
#include <hip/hip_runtime.h>
#include <math.h>

#define H 128
#define EPS 1e-5f

typedef __attribute__((ext_vector_type(2))) float v2f;
typedef __attribute__((ext_vector_type(8))) float v8f;

// ---- ordered-uint mapping for float atomic max ----
__device__ __forceinline__ unsigned mapf(float f) {
    unsigned b = __float_as_uint(f);
    return (b & 0x80000000u) ? ~b : (b | 0x80000000u);
}
__device__ __forceinline__ float unmapf(unsigned u) {
    unsigned b = (u & 0x80000000u) ? (u & 0x7FFFFFFFu) : ~u;
    return __uint_as_float(b);
}

// ---- 1) edge scores: one wave (32 lanes) per edge, lane handles 4 floats ----
__global__ void edge_score_kernel(const float* __restrict__ emb,
                                  const int* __restrict__ src,
                                  const int* __restrict__ dst,
                                  float* __restrict__ score,
                                  unsigned* __restrict__ seg_max_u,
                                  int E) {
    int gtid = blockIdx.x * blockDim.x + threadIdx.x;
    int e    = gtid >> 5;
    int lane = gtid & 31;
    if (e >= E) return;
    int s = src[e], d = dst[e];
    const float4* pa = (const float4*)(emb + (size_t)s * H);
    const float4* pb = (const float4*)(emb + (size_t)d * H);
    float4 a = pa[lane], b = pb[lane];
    float p = a.x * b.x + a.y * b.y + a.z * b.z + a.w * b.w;
    p += __shfl_xor(p, 16, 32);
    p += __shfl_xor(p,  8, 32);
    p += __shfl_xor(p,  4, 32);
    p += __shfl_xor(p,  2, 32);
    p += __shfl_xor(p,  1, 32);
    if (lane == 0) {
        score[e] = p;
        atomicMax(seg_max_u + d, mapf(p));
    }
}

// ---- 2) exp(score - max) and per-destination sum ----
__global__ void edge_exp_kernel(float* __restrict__ score,
                                const int* __restrict__ dst,
                                const unsigned* __restrict__ seg_max_u,
                                float* __restrict__ seg_sum,
                                int E) {
    int e = blockIdx.x * blockDim.x + threadIdx.x;
    if (e >= E) return;
    int d = dst[e];
    float m  = unmapf(seg_max_u[d]);
    float ex = expf(score[e] - m);
    score[e] = ex;
    atomicAdd(seg_sum + d, ex);
}

// ---- 3) weighted aggregation: one wave per edge, 4 f32 atomics per lane ----
__global__ void aggregate_kernel(const float* __restrict__ emb,
                                 const int* __restrict__ src,
                                 const int* __restrict__ dst,
                                 const float* __restrict__ ex,
                                 const float* __restrict__ seg_sum,
                                 float* __restrict__ neigh,
                                 int E) {
    int gtid = blockIdx.x * blockDim.x + threadIdx.x;
    int e    = gtid >> 5;
    int lane = gtid & 31;
    if (e >= E) return;
    int s = src[e], d = dst[e];
    float w = ex[e] / seg_sum[d];
    const float4* pa = (const float4*)(emb + (size_t)s * H);
    float4 a = pa[lane];
    float* o = neigh + (size_t)d * H + lane * 4;
    atomicAdd(o + 0, a.x * w);
    atomicAdd(o + 1, a.y * w);
    atomicAdd(o + 2, a.z * w);
    atomicAdd(o + 3, a.w * w);
}

// ---- 4) h = neigh @ W via V_WMMA_F32_16X16X4_F32 ----
// Block = 256 threads = 8 waves; wave w owns column tile [16w,16w+16),
// block b owns rows [16b, 16b+16). N must be a multiple of 16 (50000 is).
// No divergence: EXEC stays all-ones as WMMA requires.
__global__ __launch_bounds__(256) void gemm_wmma_kernel(const float* __restrict__ neigh,
                                                        const float* __restrict__ W,
                                                        float* __restrict__ out,
                                                        int N) {
    int wave = threadIdx.x >> 5;   // 0..7 -> column tile
    int lane = threadIdx.x & 31;
    int row0 = blockIdx.x * 16;
    int col0 = wave * 16;
    int half = lane >> 4;          // 0: lanes 0-15, 1: lanes 16-31
    int mn   = lane & 15;          // M for A-frag, N for B/C-frag

    v8f c = {0.f, 0.f, 0.f, 0.f, 0.f, 0.f, 0.f, 0.f};
    const float* arow = neigh + (size_t)(row0 + mn) * H;

    #pragma unroll 4
    for (int k0 = 0; k0 < H; k0 += 4) {
        int ka = k0 + half * 2;
        // A 16x4: lanes0-15 VGPR0=K0,VGPR1=K1; lanes16-31 VGPR0=K2,VGPR1=K3
        v2f a;
        a.x = arow[ka];
        a.y = arow[ka + 1];
        // B 4x16: row K striped across lanes; VGPR0: K0(lo)/K2(hi), VGPR1: K1(lo)/K3(hi)
        v2f b;
        b.x = W[(size_t)ka * H + col0 + mn];
        b.y = W[(size_t)(ka + 1) * H + col0 + mn];
        c = __builtin_amdgcn_wmma_f32_16x16x4_f32(false, a, false, b,
                                                  (short)0, c, false, false);
    }

    // C/D 16x16 f32: VGPR r -> M=r (lanes 0-15), M=r+8 (lanes 16-31)
    float* orow = out + ((size_t)row0 + (size_t)half * 8) * H + col0 + mn;
    #pragma unroll
    for (int r = 0; r < 8; ++r)
        orow[(size_t)r * H] = c[r];
}

// ---- 5) per-column sum / sum-of-squares ----
__global__ void bn_stats_kernel(const float* __restrict__ h,
                                float* __restrict__ col_sum,
                                float* __restrict__ col_sumsq,
                                int N, int rowsPerBlock) {
    int col = threadIdx.x;                 // 0..127
    int r0 = blockIdx.x * rowsPerBlock;
    int r1 = min(N, r0 + rowsPerBlock);
    float s = 0.f, ss = 0.f;
    for (int r = r0; r < r1; ++r) {
        float v = h[(size_t)r * H + col];
        s  += v;
        ss += v * v;
    }
    atomicAdd(&col_sum[col], s);
    atomicAdd(&col_sumsq[col], ss);
}

// ---- 6) fused BatchNorm (batch stats) + tanh, in place on d_out ----
__global__ void bn_tanh_kernel(float* __restrict__ h,
                               const float* __restrict__ col_sum,
                               const float* __restrict__ col_sumsq,
                               const float* __restrict__ gamma,
                               const float* __restrict__ beta,
                               int N) {
    size_t idx   = (size_t)blockIdx.x * blockDim.x + threadIdx.x;
    size_t total = (size_t)N * H;
    if (idx >= total) return;
    int col = (int)(idx & (H - 1));
    float invN = 1.0f / (float)N;
    float mu   = col_sum[col] * invN;
    float var  = col_sumsq[col] * invN - mu * mu;
    float inv  = rsqrtf(var + EPS);
    float v = (h[idx] - mu) * inv * gamma[col] + beta[col];
    h[idx] = tanhf(v);
}

extern "C" void kernel_launch(void* const* d_in, const int* in_sizes, int n_in,
                              void* d_out, int out_size, void* d_ws, size_t ws_size,
                              hipStream_t stream) {
    const float* ent   = (const float*)d_in[0];
    const float* W     = (const float*)d_in[1];
    const float* gamma = (const float*)d_in[2];
    const float* beta  = (const float*)d_in[3];
    const int*   src   = (const int*)d_in[4];
    const int*   dst   = (const int*)d_in[5];
    int N = in_sizes[0] / H;   // 50000
    int E = in_sizes[4];       // 800000
    float* out = (float*)d_out;

    // workspace layout (base is 256B aligned from hipMalloc)
    char* ws = (char*)d_ws;
    float*    neigh     = (float*)ws;    ws += (size_t)N * H * sizeof(float);
    float*    score     = (float*)ws;    ws += (size_t)E * sizeof(float);
    unsigned* seg_max_u = (unsigned*)ws; ws += (size_t)N * sizeof(unsigned);
    float*    seg_sum   = (float*)ws;    ws += (size_t)N * sizeof(float);
    float*    col_sum   = (float*)ws;    ws += H * sizeof(float);
    float*    col_sumsq = (float*)ws;    ws += H * sizeof(float);
    size_t zero_bytes = (size_t)(ws - (char*)d_ws);

    // zero accumulators (0 is also the identity for the ordered-uint max)
    hipMemsetAsync(d_ws, 0, zero_bytes, stream);

    // 1) edge scores (one wave per edge)
    {
        long long tot = (long long)E * 32;
        int blocks = (int)((tot + 255) / 256);
        edge_score_kernel<<<blocks, 256, 0, stream>>>(ent, src, dst, score, seg_max_u, E);
    }
    // 2) exp + segment sum
    edge_exp_kernel<<<(E + 255) / 256, 256, 0, stream>>>(score, dst, seg_max_u, seg_sum, E);
    // 3) weighted aggregation (one wave per edge)
    {
        long long tot = (long long)E * 32;
        int blocks = (int)((tot + 255) / 256);
        aggregate_kernel<<<blocks, 256, 0, stream>>>(ent, src, dst, score, seg_sum, neigh, E);
    }
    // 4) GEMM with f32 WMMA (N divisible by 16)
    gemm_wmma_kernel<<<N / 16, 256, 0, stream>>>(neigh, W, out, N);
    // 5) BN stats
    {
        int blocksBN = 250;
        int rowsPerBlock = (N + blocksBN - 1) / blocksBN;
        bn_stats_kernel<<<blocksBN, H, 0, stream>>>(out, col_sum, col_sumsq, N, rowsPerBlock);
    }
    // 6) fused BN + tanh
    {
        size_t tot = (size_t)N * H;
        int blocks = (int)((tot + 255) / 256);
        bn_tanh_kernel<<<blocks, 256, 0, stream>>>(out, col_sum, col_sumsq, gamma, beta, N);
    }
}